// MentionScoresHead_57337813401647
// MI455X (gfx1250) — compile-verified
//
#include <hip/hip_runtime.h>

// ---------------------------------------------------------------------------
// MentionScoresHead for MI455X (gfx1250, wave32).
// B=16, L=1024, D=768, MAX_MENTION_LEN=10.
// K1: projection GEMM via V_WMMA_F32_16X16X4_F32 (fp32 WMMA, K=4 per op).
// K2: per-batch cumsum + fused row/col score terms (LDS scan).
// K3: stream 16x533971 scores (fp32) + bounds (int64 pairs, b128 stores).
// Roofline: K1 reads 50 MB (~2.2us @ 23.3 TB/s); K3 writes ~171 MB (~7.4us)
// and dominates, so K3 is kept to 2 cached reads + 20 B written per entry.
// ---------------------------------------------------------------------------

typedef float     v2f  __attribute__((ext_vector_type(2)));
typedef float     v8f  __attribute__((ext_vector_type(8)));
typedef long long v2ll __attribute__((ext_vector_type(2)));

#define BB   16
#define LL   1024
#define DD   768
// kept pairs per batch: sum_{s=0}^{1023} min(s+10, 1024)
#define KTOT   533971
// prefix P(s) = s*(s+19)/2 valid for s <= 1015; P(1015) = 524755
#define P1015  524755

__device__ __forceinline__ int prefixP(int s) { return (s * (s + 19)) >> 1; }

// ---------------------------------------------------------------------------
// Kernel 1: logits[b,l,0:3] = bert[b,l,:] . W[k,:] + bias[k]  via fp32 WMMA.
// One wave per 16-token tile; A: 16x4 (M=token), B: 4x16 (N=output, cols>=3
// zeroed by cndmask so EXEC stays full for the WMMA). 192 WMMAs per tile.
// A-layout (ISA 7.12.2): lane m (0..15), half h (lane>>4):
//   vgpr0 = A[m, k0+2h], vgpr1 = A[m, k0+2h+1]; B mirrored.
// D-layout: vgpr r, lane n (0..15), half h -> D[M=r+8h, N=n].
// ---------------------------------------------------------------------------
__global__ __launch_bounds__(256) void proj_wmma_kernel(
    const float* __restrict__ bert,   // [B, L, D]
    const float* __restrict__ W,      // [3, D]
    const float* __restrict__ bias,   // [3]
    float* __restrict__ logits)       // [B, L, 3]
{
  const int lane = threadIdx.x & 31;
  const int wave = threadIdx.x >> 5;
  const int tile = blockIdx.x * 8 + wave;      // 1024 tiles total (B*L/16)
  const int b    = tile >> 6;                  // 64 tiles per batch
  const int rowBase = (tile & 63) << 4;        // token base within batch
  const int mn = lane & 15;                    // A row M / B col N
  const int kh = lane >> 4;                    // K-half selector

  const float* aPtr = bert + ((size_t)b * LL + rowBase + mn) * DD;
  const int   wn    = (mn < 3) ? mn : 0;       // clamp row -> no OOB read
  const float* wPtr = W + (size_t)wn * DD;
  const bool  bLive = (mn < 3);

  v8f acc = {};
  #pragma unroll 4
  for (int k0 = 0; k0 < DD; k0 += 4) {
    v2f a = *(const v2f*)(aPtr + k0 + 2 * kh);
    v2f w = *(const v2f*)(wPtr + k0 + 2 * kh);
    if (!bLive) { w = (v2f){0.0f, 0.0f}; }     // v_cndmask, EXEC unchanged
    acc = __builtin_amdgcn_wmma_f32_16x16x4_f32(
        /*neg_a=*/false, a, /*neg_b=*/false, w,
        /*c_mod=*/(short)0, acc, /*reuse_a=*/false, /*reuse_b=*/false);
  }

  if (bLive) {
    const float bv = bias[mn];
    float* out = logits + ((size_t)b * LL + rowBase + 8 * kh) * 3 + mn;
    #pragma unroll
    for (int r = 0; r < 8; ++r)
      out[(size_t)r * 3] = acc[r] + bv;        // D[M=r+8*kh, N=mn]
  }
}

// ---------------------------------------------------------------------------
// Kernel 2: per-batch LDS scan of mention logits, then fused score terms:
//   incl[t]    = cumsum(logit2[0..t])                  (raw; always finite)
//   rowterm[t] = logit0[t] - (t>0 ? incl[t-1] : 0)     (start - excl)
//   colterm[t] = mask[t] ? logit1[t] + incl[t] : -inf  (end + incl, masked)
// score(s,e) = rowterm[s] + colterm[e] for e>=s; -inf propagation via colterm
// matches the reference (valid requires mask[e]; prefix mask => s,e finite).
// ---------------------------------------------------------------------------
__global__ __launch_bounds__(1024) void terms_kernel(
    const float* __restrict__ logits,         // [B, L, 3]
    const unsigned char* __restrict__ mask,   // [B, L] bool
    float* __restrict__ rowterm,              // [B, L]
    float* __restrict__ colterm)              // [B, L]
{
  __shared__ float sm[LL];
  const int b = blockIdx.x;
  const int t = threadIdx.x;
  const float* lg = logits + ((size_t)b * LL + t) * 3;
  const float l0 = lg[0], l1 = lg[1], l2 = lg[2];
  sm[t] = l2;
  __syncthreads();
  #pragma unroll
  for (int off = 1; off < LL; off <<= 1) {
    float add = (t >= off) ? sm[t - off] : 0.0f;
    __syncthreads();
    sm[t] += add;
    __syncthreads();
  }
  const float inc = sm[t];
  const float exc = (t > 0) ? sm[t - 1] : 0.0f;
  rowterm[(size_t)b * LL + t] = l0 - exc;
  colterm[(size_t)b * LL + t] =
      mask[(size_t)b * LL + t] ? (l1 + inc) : -__builtin_inff();
}

// ---------------------------------------------------------------------------
// Kernel 3: one thread per kept (s,e) pair; grid = (ceil(KTOT/256), B).
// kept order = row-major (s,e) with e in [0, min(s+9, 1023)].
// ---------------------------------------------------------------------------
__global__ __launch_bounds__(256) void scores_kernel(
    const float* __restrict__ rowterm,        // [B, L]
    const float* __restrict__ colterm,        // [B, L]
    float* __restrict__ scores,               // [B, KTOT]
    v2ll* __restrict__ bounds)                // [B, KTOT] of (s, e)
{
  const int idx = blockIdx.x * 256 + threadIdx.x;
  if (idx >= KTOT) return;
  const int b = blockIdx.y;

  int s, e;
  if (idx >= P1015) {                          // linear tail: 1024 cols/row
    const int r = idx - P1015;
    s = 1015 + (r >> 10);
    e = r & 1023;
  } else {                                     // invert P(s) = s(s+19)/2
    const float f = sqrtf(8.0f * (float)idx + 361.0f);
    int sg = (int)((f - 19.0f) * 0.5f);
    if (sg < 0) sg = 0;
    if (sg > 1015) sg = 1015;
    while (sg < 1015 && prefixP(sg + 1) <= idx) ++sg;
    while (sg > 0 && prefixP(sg) > idx) --sg;
    s = sg;
    e = idx - prefixP(sg);
  }

  const float sc = (e >= s)
      ? rowterm[(size_t)b * LL + s] + colterm[(size_t)b * LL + e]
      : -__builtin_inff();

  const size_t o = (size_t)b * KTOT + idx;
  scores[o] = sc;
  bounds[o] = (v2ll){(long long)s, (long long)e};  // single b128 store
}

// ---------------------------------------------------------------------------
extern "C" void kernel_launch(void* const* d_in, const int* in_sizes, int n_in,
                              void* d_out, int out_size, void* d_ws, size_t ws_size,
                              hipStream_t stream) {
  (void)in_sizes; (void)n_in; (void)out_size; (void)ws_size;

  const float*         bert = (const float*)d_in[0];          // [16,1024,768]
  const unsigned char* mask = (const unsigned char*)d_in[1];  // [16,1024] bool
  const float*         W    = (const float*)d_in[2];          // [3,768]
  const float*         bias = (const float*)d_in[3];          // [3]

  // scratch: logits [B*L*3] | rowterm [B*L] | colterm [B*L]  (~320 KB)
  float* logits  = (float*)d_ws;
  float* rowterm = logits + (size_t)BB * LL * 3;
  float* colterm = rowterm + (size_t)BB * LL;

  // output: scores fp32 [B*KTOT], then bounds int64 [B*KTOT*2].
  // bounds byte offset 16*KTOT*4 = 34,174,144 is 16B-aligned -> b128 stores.
  float* scores = (float*)d_out;
  v2ll*  bounds = (v2ll*)((char*)d_out + (size_t)BB * KTOT * sizeof(float));

  proj_wmma_kernel<<<128, 256, 0, stream>>>(bert, W, bias, logits);
  terms_kernel<<<BB, LL, 0, stream>>>(logits, mask, rowterm, colterm);

  dim3 grid3((KTOT + 255) / 256, BB);
  scores_kernel<<<grid3, 256, 0, stream>>>(rowterm, colterm, scores, bounds);
}